// InteractionBlock_39573828666265
// MI455X (gfx1250) — compile-verified
//
#include <hip/hip_runtime.h>
#include <hip/hip_bf16.h>

typedef float v2f __attribute__((ext_vector_type(2)));
typedef float v4f __attribute__((ext_vector_type(4)));
typedef float v8f __attribute__((ext_vector_type(8)));

// D(16x16,f32) = A(16x4,f32) * B(4x16,f32) + C
static __device__ __forceinline__ v8f wmma4(v2f a, v2f b, v8f c) {
  return __builtin_amdgcn_wmma_f32_16x16x4_f32(
      /*neg_a=*/false, a, /*neg_b=*/false, b,
      /*c_mod=*/(short)0, c, /*reuse_a=*/false, /*reuse_b=*/false);
}

// ---------------------------------------------------------------- zero scratch
__global__ void k_zero(v4f* __restrict__ p, long nvec) {
  long i = (long)blockIdx.x * blockDim.x + threadIdx.x;
  if (i < nvec) {
    v4f z = {0.f, 0.f, 0.f, 0.f};
    p[i] = z;
  }
}

// ---------------------------------------------------------------- lin1 (WMMA)
// ys[n][w] = (xs @ W1s)[n][w] / sqrt(32);  yv[n][w][i] = (xv_i @ W1v)[n][w] / sqrt(32)
__global__ __launch_bounds__(32) void k_lin1(
    const float* __restrict__ x,     // (n,128): [:,:32]=xs, [:,32:]=xv (u,i)
    const float* __restrict__ W1s,   // (32,32)
    const float* __restrict__ W1v,   // (32,32)
    float* __restrict__ ys,          // (n,32)
    float* __restrict__ yv,          // (n,32,3)
    int n) {
  const int lane  = threadIdx.x & 31;
  const int base  = blockIdx.x * 16;
  const int half  = lane >> 4;       // 0 or 1
  const int l15   = lane & 15;
  int row = base + l15;
  if (row >= n) row = n - 1;         // tail clamp (loads only)
  const int khalf = half * 2;

  v8f acc[8] = {};                   // [mat: s,v0,v1,v2][ntile 0..1]

#pragma unroll
  for (int k0 = 0; k0 < 32; k0 += 4) {
    const int kk = k0 + khalf;
    v2f a_s;
    a_s[0] = x[row * 128 + kk];
    a_s[1] = x[row * 128 + kk + 1];
    v2f a_v[3];
#pragma unroll
    for (int i = 0; i < 3; ++i) {
      a_v[i][0] = x[row * 128 + 32 + kk * 3 + i];
      a_v[i][1] = x[row * 128 + 32 + (kk + 1) * 3 + i];
    }
#pragma unroll
    for (int t = 0; t < 2; ++t) {
      const int c = t * 16 + l15;
      v2f b_s, b_v;
      b_s[0] = W1s[kk * 32 + c];
      b_s[1] = W1s[(kk + 1) * 32 + c];
      b_v[0] = W1v[kk * 32 + c];
      b_v[1] = W1v[(kk + 1) * 32 + c];
      acc[t] = wmma4(a_s, b_s, acc[t]);
#pragma unroll
      for (int i = 0; i < 3; ++i)
        acc[(1 + i) * 2 + t] = wmma4(a_v[i], b_v, acc[(1 + i) * 2 + t]);
    }
  }

  const float l1n = 0.17677669529663687f;  // 1/sqrt(32)
  const int mbase = base + half * 8;
#pragma unroll
  for (int r = 0; r < 8; ++r) {
    const int m = mbase + r;
    if (m >= n) break;
#pragma unroll
    for (int t = 0; t < 2; ++t) {
      const int c = t * 16 + l15;
      ys[(size_t)m * 32 + c]         = acc[t][r] * l1n;
      yv[(size_t)m * 96 + c * 3 + 0] = acc[2 + t][r] * l1n;
      yv[(size_t)m * 96 + c * 3 + 1] = acc[4 + t][r] * l1n;
      yv[(size_t)m * 96 + c * 3 + 2] = acc[6 + t][r] * l1n;
    }
  }
}

// ---------------------------------------------------------------- edge scatter
// Streamed edge inputs use non-temporal loads (keep 192MB L2 for ys/yv + mid).
// Gathers issued as b128 (v4f) chunks.
__global__ __launch_bounds__(256) void k_edge(
    const float* __restrict__ emb,   // (ne,8)
    const float* __restrict__ ea,    // (ne,4)
    const int*   __restrict__ eidx,  // (2,ne): [0]=src, [1]=dst
    const float* __restrict__ fc1,   // (8,8)   uniform -> SMEM
    const float* __restrict__ fc2,   // (8,128) uniform -> SMEM
    const float* __restrict__ ys,    // (n,32)
    const float* __restrict__ yv,    // (n,32,3)
    float* __restrict__ mids,        // (n,64)   unscaled segment sums
    float* __restrict__ midv,        // (n,64,3) unscaled segment sums
    int ne) {
  const int e = blockIdx.x * blockDim.x + threadIdx.x;
  if (e >= ne) return;

  const float isq8 = 0.35355339059327373f;   // 1/sqrt(8)

  // edge embedding: 2 x b128, non-temporal (streamed once)
  const v4f* emb4 = (const v4f*)(emb + (size_t)e * 8);
  const v4f em0 = __builtin_nontemporal_load(emb4 + 0);
  const v4f em1 = __builtin_nontemporal_load(emb4 + 1);
  const float em[8] = {em0[0], em0[1], em0[2], em0[3],
                       em1[0], em1[1], em1[2], em1[3]};

  float h[8];
#pragma unroll
  for (int j = 0; j < 8; ++j) {
    float s = 0.0f;
#pragma unroll
    for (int k = 0; k < 8; ++k) s += em[k] * fc1[k * 8 + j];
    s *= isq8;
    h[j] = s / (1.0f + __expf(-s));          // silu
  }

  const int src = __builtin_nontemporal_load(eidx + e);
  const int dst = __builtin_nontemporal_load(eidx + ne + e);
  const v4f av = __builtin_nontemporal_load((const v4f*)(ea + (size_t)e * 4));
  const float a0 = av[0], a1x = av[1], a1y = av[2], a1z = av[3];
  const float isq3 = 0.5773502691896258f;    // 1/sqrt(3)

  float* ms = mids + (size_t)dst * 64;
  float* mv = midv + (size_t)dst * 192;
  const float* es_p = ys + (size_t)src * 32;
  const float* ev_p = yv + (size_t)src * 96;

  for (int uc = 0; uc < 32; uc += 4) {
    // gather chunk: 1 x b128 of es, 3 x b128 of ev  (rows are 16B aligned)
    const v4f esv = *(const v4f*)(es_p + uc);
    float evl[12];
    *(v4f*)(evl + 0) = *(const v4f*)(ev_p + uc * 3);
    *(v4f*)(evl + 4) = *(const v4f*)(ev_p + uc * 3 + 4);
    *(v4f*)(evl + 8) = *(const v4f*)(ev_p + uc * 3 + 8);

#pragma unroll
    for (int du = 0; du < 4; ++du) {
      const int u = uc + du;
      float w00 = 0.f, w01 = 0.f, w10 = 0.f, w11 = 0.f;
#pragma unroll
      for (int k = 0; k < 8; ++k) {
        const float* f2 = fc2 + k * 128;   // wave-uniform -> scalar loads
        w00 += h[k] * f2[u];
        w01 += h[k] * f2[32 + u];
        w10 += h[k] * f2[64 + u];
        w11 += h[k] * f2[96 + u];
      }
      w00 *= isq8; w01 *= isq8; w10 *= isq8; w11 *= isq8;

      const float es  = esv[du];
      const float evx = evl[du * 3 + 0];
      const float evy = evl[du * 3 + 1];
      const float evz = evl[du * 3 + 2];

      atomicAdd(ms + u,      w00 * es * a0);
      atomicAdd(ms + 32 + u, w11 * (evx * a1x + evy * a1y + evz * a1z) * isq3);

      const float c01 = w01 * es;
      atomicAdd(mv + u * 3 + 0, c01 * a1x);
      atomicAdd(mv + u * 3 + 1, c01 * a1y);
      atomicAdd(mv + u * 3 + 2, c01 * a1z);

      const float c10 = w10 * a0;
      atomicAdd(mv + 96 + u * 3 + 0, c10 * evx);
      atomicAdd(mv + 96 + u * 3 + 1, c10 * evy);
      atomicAdd(mv + 96 + u * 3 + 2, c10 * evz);
    }
  }
}

// ---------------------------------------------------------------- lin2 + sc (WMMA)
__global__ __launch_bounds__(32) void k_lin2(
    const float* __restrict__ x,     // (n,128) original node features
    const float* __restrict__ attr,  // (n,4) one-hot species
    const float* __restrict__ W2s,   // (64,32)
    const float* __restrict__ W2v,   // (64,32)
    const float* __restrict__ Wscs,  // (32,4,32)
    const float* __restrict__ Wscv,  // (32,4,32)
    const float* __restrict__ mids,  // (n,64)
    const float* __restrict__ midv,  // (n,64,3)
    float* __restrict__ out,         // (n,128)
    int n) {
  const int lane  = threadIdx.x & 31;
  const int base  = blockIdx.x * 16;
  const int half  = lane >> 4;
  const int l15   = lane & 15;
  int row = base + l15;
  if (row >= n) row = n - 1;
  const int khalf = half * 2;

  v8f acc[8] = {};

  for (int k0 = 0; k0 < 64; k0 += 4) {
    const int kk = k0 + khalf;
    v2f a_s;
    a_s[0] = mids[(size_t)row * 64 + kk];
    a_s[1] = mids[(size_t)row * 64 + kk + 1];
    v2f a_v[3];
#pragma unroll
    for (int i = 0; i < 3; ++i) {
      a_v[i][0] = midv[(size_t)row * 192 + kk * 3 + i];
      a_v[i][1] = midv[(size_t)row * 192 + (kk + 1) * 3 + i];
    }
#pragma unroll
    for (int t = 0; t < 2; ++t) {
      const int c = t * 16 + l15;
      v2f b_s, b_v;
      b_s[0] = W2s[kk * 32 + c];
      b_s[1] = W2s[(kk + 1) * 32 + c];
      b_v[0] = W2v[kk * 32 + c];
      b_v[1] = W2v[(kk + 1) * 32 + c];
      acc[t] = wmma4(a_s, b_s, acc[t]);
#pragma unroll
      for (int i = 0; i < 3; ++i)
        acc[(1 + i) * 2 + t] = wmma4(a_v[i], b_v, acc[(1 + i) * 2 + t]);
    }
  }

  const float scale = 0.03125f;              // (1/sqrt(16)) * (1/sqrt(64))
  const float scn   = 0.08838834764831845f;  // 1/sqrt(32*4)
  const int mbase = base + half * 8;
#pragma unroll
  for (int r = 0; r < 8; ++r) {
    const int m = mbase + r;
    if (m >= n) break;
    int sp = 0;
#pragma unroll
    for (int v = 1; v < 4; ++v)
      if (attr[(size_t)m * 4 + v] > 0.5f) sp = v;
    const float* Wss = Wscs + sp * 32;
    const float* Wsv = Wscv + sp * 32;
#pragma unroll
    for (int t = 0; t < 2; ++t) {
      const int c = t * 16 + l15;
      float scs = 0.f, sv0 = 0.f, sv1 = 0.f, sv2 = 0.f;
      for (int u = 0; u < 32; ++u) {
        const float ws_e = Wss[u * 128 + c];
        const float wv_e = Wsv[u * 128 + c];
        scs += x[(size_t)m * 128 + u] * ws_e;
        sv0 += x[(size_t)m * 128 + 32 + u * 3 + 0] * wv_e;
        sv1 += x[(size_t)m * 128 + 32 + u * 3 + 1] * wv_e;
        sv2 += x[(size_t)m * 128 + 32 + u * 3 + 2] * wv_e;
      }
      out[(size_t)m * 128 + c]              = acc[t][r] * scale + scs * scn;
      out[(size_t)m * 128 + 32 + c * 3 + 0] = acc[2 + t][r] * scale + sv0 * scn;
      out[(size_t)m * 128 + 32 + c * 3 + 1] = acc[4 + t][r] * scale + sv1 * scn;
      out[(size_t)m * 128 + 32 + c * 3 + 2] = acc[6 + t][r] * scale + sv2 * scn;
    }
  }
}

// ---------------------------------------------------------------- launch
extern "C" void kernel_launch(void* const* d_in, const int* in_sizes, int n_in,
                              void* d_out, int out_size, void* d_ws, size_t ws_size,
                              hipStream_t stream) {
  const float* x    = (const float*)d_in[0];   // node_features (n,128)
  const float* attr = (const float*)d_in[1];   // node_attr (n,4)
  const float* emb  = (const float*)d_in[2];   // edge_embedding (ne,8)
  const float* ea   = (const float*)d_in[3];   // edge_attr (ne,4)
  const int*   eidx = (const int*)d_in[4];     // edge_index (2,ne)
  const float* W1s  = (const float*)d_in[5];
  const float* W1v  = (const float*)d_in[6];
  const float* fc1  = (const float*)d_in[7];
  const float* fc2  = (const float*)d_in[8];
  const float* W2s  = (const float*)d_in[9];
  const float* W2v  = (const float*)d_in[10];
  const float* Wscs = (const float*)d_in[11];
  const float* Wscv = (const float*)d_in[12];
  float* out = (float*)d_out;

  const int n  = in_sizes[0] / 128;
  const int ne = in_sizes[2] / 8;

  // scratch layout (floats): ys 32n | yv 96n | mid_s 64n | mid_v 192n  = 384n
  float* ws   = (float*)d_ws;
  float* ys   = ws;
  float* yv   = ws + (size_t)32 * n;
  float* mids = ws + (size_t)128 * n;
  float* midv = ws + (size_t)192 * n;

  const long zvec = (long)n * 64;   // (mid_s + mid_v) = 256n floats = 64n v4f
  k_zero<<<(int)((zvec + 255) / 256), 256, 0, stream>>>((v4f*)mids, zvec);

  const int tiles = (n + 15) / 16;
  k_lin1<<<tiles, 32, 0, stream>>>(x, W1s, W1v, ys, yv, n);
  k_edge<<<(ne + 255) / 256, 256, 0, stream>>>(emb, ea, eidx, fc1, fc2,
                                               ys, yv, mids, midv, ne);
  k_lin2<<<tiles, 32, 0, stream>>>(x, attr, W2s, W2v, Wscs, Wscv,
                                   mids, midv, out, n);
}